// RotationHopfieldNetwork2D_34230889349748
// MI455X (gfx1250) — compile-verified
//
#include <hip/hip_runtime.h>
#include <hip/hip_bf16.h>
#include <stdint.h>

// CDNA5 wave32 WMMA operand types
typedef __attribute__((ext_vector_type(2))) float v2f;   // A/B of V_WMMA_F32_16X16X4_F32
typedef __attribute__((ext_vector_type(8))) float v8f;   // 16x16 f32 C/D

#define PATS   512
#define NCHUNK (PATS / 16)     // 32 WMMA chunks of 16 patterns
#define BETA_C 0.01f
#define LR_C   0.01f

__global__ __launch_bounds__(256)
void hopfield2d_wmma_kernel(const float*  __restrict__ x_in,
                            const float*  __restrict__ patterns,
                            const unsigned char* __restrict__ mask,
                            const int*    __restrict__ max_iter_p,
                            float*        __restrict__ x_out,
                            int ntiles)
{
    // phi(patterns) staged once per block: 512 * float4 = 8 KB LDS.
    // float4 packing -> one conflict-free ds_load_b128 per lane per chunk
    // (lane l reads dwords 4l..4l+3: 64 banks covered once per 16 lanes).
    __shared__ float4 spp[PATS];

    const int tid = threadIdx.x;
    for (int j = tid; j < PATS; j += blockDim.x) {
        float a = patterns[2 * j], b = patterns[2 * j + 1];
        float a2 = a * a, b2 = b * b;
        spp[j] = make_float4(a2 + b2,            // f1
                             a2 * b2,            // f2
                             a * b * (a2 - b2),  // f3 = a^3 b - a b^3
                             0.0f);
    }
    __syncthreads();

    const int wave = tid >> 5;
    const int lane = tid & 31;
    const int tile = blockIdx.x * (blockDim.x >> 5) + wave;
    if (tile >= ntiles) return;                 // uniform per wave -> EXEC all-ones below
    const int  base     = tile * 16;
    const int  l15      = lane & 15;
    const int  r8       = lane & 7;
    const bool lowhalf  = (lane < 16);
    const int  mysample = base + l15;           // sample whose x this lane carries
    // canonical lane holding sample (lane&15) after the update step:
    const int  src_lane = (l15 < 8) ? l15 : (l15 + 16);

    // x replicated across both 16-lane halves (A operand needs rows on both halves)
    float xa = x_in[2 * mysample];
    float xb = x_in[2 * mysample + 1];
    const bool ma = mask[2 * mysample]     != 0;
    const bool mb = mask[2 * mysample + 1] != 0;

    const int niter = max_iter_p[0];

    for (int it = 0; it < niter; ++it) {
        // ---- build A (16x4 f32, K = {f1,f2,f3,0}) per ISA layout:
        // lanes 0-15: VGPR0=K0, VGPR1=K1 ; lanes 16-31: VGPR0=K2, VGPR1=K3
        float a2 = xa * xa, b2 = xb * xb;
        float f1 = a2 + b2;
        float f2 = a2 * b2;
        float f3 = xa * xb * (a2 - b2);
        v2f Amat;
        Amat.x = lowhalf ? f1 : f3;
        Amat.y = lowhalf ? f2 : 0.0f;

        // ================= PASS 1: row max of raw logits (beta folded in later)
        float m[8];
        #pragma unroll
        for (int r = 0; r < 8; ++r) m[r] = -1e30f;

        #pragma unroll 4
        for (int t = 0; t < NCHUNK; ++t) {
            const float4 p = spp[t * 16 + l15];
            v2f Bmat;                            // B (4x16): col N = lane&15
            Bmat.x = lowhalf ? p.x : p.z;        // K0 / K2
            Bmat.y = lowhalf ? p.y : 0.0f;       // K1 / K3
            v8f C = {};
            C = __builtin_amdgcn_wmma_f32_16x16x4_f32(
                    false, Amat, false, Bmat, (short)0, C, false, false);
            #pragma unroll
            for (int r = 0; r < 8; ++r) m[r] = fmaxf(m[r], C[r]);
        }
        // butterfly max-reduce within each 16-lane half (C-layout sample split)
        #pragma unroll
        for (int mk = 1; mk <= 8; mk <<= 1) {
            #pragma unroll
            for (int r = 0; r < 8; ++r)
                m[r] = fmaxf(m[r], __shfl_xor(m[r], mk, 32));
        }
        float M[8];
        #pragma unroll
        for (int r = 0; r < 8; ++r) M[r] = BETA_C * m[r];   // max of beta-scaled logits

        // ================= PASS 2: exp-weighted sums with common max (1 exp/logit)
        float S[8], A1[8], A2[8], A3[8];
        #pragma unroll
        for (int r = 0; r < 8; ++r) { S[r] = 0.f; A1[r] = 0.f; A2[r] = 0.f; A3[r] = 0.f; }

        #pragma unroll 4
        for (int t = 0; t < NCHUNK; ++t) {
            const float4 p = spp[t * 16 + l15];
            v2f Bmat;
            Bmat.x = lowhalf ? p.x : p.z;
            Bmat.y = lowhalf ? p.y : 0.0f;
            v8f C = {};
            C = __builtin_amdgcn_wmma_f32_16x16x4_f32(
                    false, Amat, false, Bmat, (short)0, C, false, false);
            #pragma unroll
            for (int r = 0; r < 8; ++r) {
                float w = __expf(fmaf(BETA_C, C[r], -M[r]));  // <= 1, no overflow
                S[r]  += w;
                A1[r] += w * p.x;
                A2[r] += w * p.y;
                A3[r] += w * p.z;
            }
        }

        // butterfly plain-sum merge (no exps needed: common max)
        #pragma unroll
        for (int mk = 1; mk <= 8; mk <<= 1) {
            #pragma unroll
            for (int r = 0; r < 8; ++r) {
                S[r]  += __shfl_xor(S[r],  mk, 32);
                A1[r] += __shfl_xor(A1[r], mk, 32);
                A2[r] += __shfl_xor(A2[r], mk, 32);
                A3[r] += __shfl_xor(A3[r], mk, 32);
            }
        }

        // ---- gradient + update (valid on canonical lanes {0-7, 24-31})
        float invS = 1.0f / S[r8];
        float g1 = f1 - A1[r8] * invS;
        float g2 = f2 - A2[r8] * invS;
        float g3 = f3 - A3[r8] * invS;
        // J^T g : Jacobian of phi wrt (a,b)
        float ga = 2.f * xa * g1 + 2.f * xa * b2 * g2 + (3.f * a2 * xb - xb * b2) * g3;
        float gb = 2.f * xb * g1 + 2.f * a2 * xb * g2 + (xa * a2 - 3.f * xa * b2) * g3;
        float xan = ma ? (xa - LR_C * ga) : xa;
        float xbn = mb ? (xb - LR_C * gb) : xb;

        // redistribute canonical results so every lane again holds x[lane&15]
        xa = __shfl(xan, src_lane, 32);
        xb = __shfl(xbn, src_lane, 32);
    }

    // after redistribution all lanes hold the right value; lanes 0-15 write the tile
    if (lane < 16) {
        const int o = base + lane;
        x_out[2 * o]     = xa;
        x_out[2 * o + 1] = xb;
    }
}

extern "C" void kernel_launch(void* const* d_in, const int* in_sizes, int n_in,
                              void* d_out, int out_size, void* d_ws, size_t ws_size,
                              hipStream_t stream) {
    const float*         x        = (const float*)d_in[0];          // (B,2) f32
    const float*         patterns = (const float*)d_in[1];          // (512,2) f32
    const unsigned char* mask     = (const unsigned char*)d_in[2];  // (B,2) bool
    const int*           max_iter = (const int*)d_in[3];            // scalar
    float*               out      = (float*)d_out;                  // (B,2) f32

    const int nsamples = in_sizes[0] / 2;
    const int ntiles   = nsamples / 16;          // one wave32 per 16 samples
    const int threads  = 256;                    // 8 waves -> 128 samples per block
    const int wavesPerBlock = threads / 32;
    const int blocks   = (ntiles + wavesPerBlock - 1) / wavesPerBlock;

    hopfield2d_wmma_kernel<<<blocks, threads, 0, stream>>>(
        x, patterns, mask, max_iter, out, ntiles);
}